// ResonantKernel_45964740002431
// MI455X (gfx1250) — compile-verified
//
#include <hip/hip_runtime.h>
#include <math.h>

typedef __attribute__((ext_vector_type(2))) float v2f;
typedef __attribute__((ext_vector_type(8))) float v8f;

#define NX 9   // 4*DIM_Q+1 state floats per row
#define NU 2   // DIM_Q

// Cross-half swap within wave32: ds_swizzle SWAPX16 (xor=0x10, and=0x1f).
__device__ __forceinline__ float swz16(float x) {
    return __int_as_float(__builtin_amdgcn_ds_swizzle(__float_as_int(x), 0x401F));
}

// Branch-free tanh: prefer the CDNA5 hardware V_TANH_F32.
__device__ __forceinline__ float fast_tanh(float x) {
#if __has_builtin(__builtin_amdgcn_tanhf)
    return __builtin_amdgcn_tanhf(x);
#else
    // tanh(x) = 1 - 2/(exp(2x)+1); exp via v_exp_f32 (base-2), v_rcp_f32.
    float e = __builtin_amdgcn_exp2f(x * 2.8853900817779268f); // 2*log2(e)
    float r = __builtin_amdgcn_rcpf(e + 1.0f);
    return __builtin_fmaf(-2.0f, r, 1.0f);
#endif
}

// One lane owns one batch row. Each wave processes 32 rows with two
// V_WMMA_F32_16X16X4_F32 ops computing PRE^T = W1 x q^T + b1 (exact fp32,
// K=2 real + 2 zero-padded). D-layout puts hidden units j=0..7 of row m in
// lane m, j=8..15 in lane m+16 -> one ds_swizzle finishes each reduction.
__global__ __launch_bounds__(256) void resonant_step_kernel(
    const float* __restrict__ x_int, const float* __restrict__ u_t,
    const float* __restrict__ W1,    const float* __restrict__ b1,
    const float* __restrict__ W2,    const float* __restrict__ b2,
    float* __restrict__ out, int batch)
{
    const int tid  = blockIdx.x * blockDim.x + threadIdx.x;
    const int lane = threadIdx.x & 31;
    const int R    = (tid >> 5) << 5;          // first row of this wave
    if (R >= batch) return;                    // wave-uniform: EXEC stays full

    const bool hi    = lane >= 16;
    const int  jbase = hi ? 8 : 0;
    const int  m     = lane & 15;

    // ---- loop-invariant operands -------------------------------------
    // A (16x4 f32): lanes 0-15 hold M=lane with K=0/1; lanes 16-31 are the
    // K=2/3 rows which we zero-pad.
    v2f A;
    {
        float w10 = W1[m * 2 + 0];
        float w11 = W1[m * 2 + 1];
        A.x = hi ? 0.0f : w10;
        A.y = hi ? 0.0f : w11;
    }
    const float b2v = b2[0];

    // Per-D-slot constants: slot g corresponds to hidden unit j = g + jbase.
    float cb1[8], cw2[8], cg0[8], cg1[8];
#pragma unroll
    for (int g = 0; g < 8; ++g) {
        int   j  = g + jbase;
        float w2 = W2[j];
        cb1[g] = b1[j];
        cw2[g] = w2;
        cg0[g] = w2 * W1[j * 2 + 0];
        cg1[g] = w2 * W1[j * 2 + 1];
    }

    // ---- load this lane's full row (clamped so EXEC stays full) ------
    int row  = R + lane;
    int rowc = row < batch ? row : (batch - 1);
    const float* xr = x_int + (long)rowc * NX;
    float q0 = __builtin_nontemporal_load(xr + 0);
    float q1 = __builtin_nontemporal_load(xr + 1);
    float p0 = __builtin_nontemporal_load(xr + 2);
    float p1 = __builtin_nontemporal_load(xr + 3);
    float s  = __builtin_nontemporal_load(xr + 4);
    float r0 = __builtin_nontemporal_load(xr + 5);
    float r1 = __builtin_nontemporal_load(xr + 6);
    float v0 = __builtin_nontemporal_load(xr + 7);
    float v1 = __builtin_nontemporal_load(xr + 8);
    const float* ur = u_t + (long)rowc * NU;
    float u0 = __builtin_nontemporal_load(ur + 0);
    float u1 = __builtin_nontemporal_load(ur + 1);

    // ---- B operands (4x16 f32): lanes 0-15 = K rows 0/1, lanes 16-31 pad 0
    v2f B0, B1;
    B0.x = hi ? 0.0f : q0;                 // block A: rows R..R+15
    B0.y = hi ? 0.0f : q1;
    float q0o = swz16(q0);                 // q of row (lane^16)
    float q1o = swz16(q1);
    B1.x = hi ? 0.0f : q0o;                // block B: rows R+16..R+31
    B1.y = hi ? 0.0f : q1o;

    v8f C;
#pragma unroll
    for (int g = 0; g < 8; ++g) C[g] = cb1[g];   // C[j,m] = b1[j]

    // D = A(W1) x B(q^T) + C(b1)  -> PRE^T, exact fp32
    v8f PT0 = __builtin_amdgcn_wmma_f32_16x16x4_f32(
        false, A, false, B0, (short)0, C, false, false);
    v8f PT1 = __builtin_amdgcn_wmma_f32_16x16x4_f32(
        false, A, false, B1, (short)0, C, false, false);

    // ---- hidden-layer epilogue: 8 units per lane per block ------------
    float mlpA = 0.f, gA0 = 0.f, gA1 = 0.f;   // rows R..R+15
    float mlpB = 0.f, gB0 = 0.f, gB1 = 0.f;   // rows R+16..R+31
#pragma unroll
    for (int g = 0; g < 8; ++g) {
        float h0 = fast_tanh(PT0[g]);
        float t0 = __builtin_fmaf(-h0, h0, 1.0f);
        mlpA = __builtin_fmaf(cw2[g], h0, mlpA);
        gA0  = __builtin_fmaf(cg0[g], t0, gA0);
        gA1  = __builtin_fmaf(cg1[g], t0, gA1);
        float h1 = fast_tanh(PT1[g]);
        float t1 = __builtin_fmaf(-h1, h1, 1.0f);
        mlpB = __builtin_fmaf(cw2[g], h1, mlpB);
        gB0  = __builtin_fmaf(cg0[g], t1, gB0);
        gB1  = __builtin_fmaf(cg1[g], t1, gB1);
    }

    // Each lane needs (own-block partial) + (partner's partial of the SAME
    // block). Partner of a lo lane is hi (block B) and vice versa, so send
    // the partial the partner needs and keep our own: 1 swizzle per quantity.
    float mlps = (hi ? mlpB : mlpA) + swz16(hi ? mlpA : mlpB);
    float gq0  = (hi ? gB0  : gA0 ) + swz16(hi ? gA0  : gB0 );
    float gq1  = (hi ? gB1  : gA1 ) + swz16(hi ? gA1  : gB1 );

    // ---- integration (per-lane scalar fp32) --------------------------
    const float DT = 0.1f, DAMP = 0.1f, TZO = 0.2f; // 2*zeta*omega
    float qn   = q0 * q0 + q1 * q1;
    float rel  = fmaxf(qn - 20.0f, 0.0f);
    float pot  = mlps + b2v + rel * rel;            // potential(q)
    float grd0 = __builtin_fmaf(4.0f * rel, q0, gq0); // grad_q
    float grd1 = __builtin_fmaf(4.0f * rel, q1, gq1);

    float pn0 = p0 - (grd0 + p0 * DAMP) * DT + u0 * DT;
    float pn1 = p1 - (grd1 + p1 * DAMP) * DT + u1 * DT;
    float qn0 = __builtin_fmaf(pn0, DT, q0);
    float qn1 = __builtin_fmaf(pn1, DT, q1);

    float psq = p0 * p0 + p1 * p1;
    float H   = 0.5f * psq + pot + DAMP * s;
    float sn  = __builtin_fmaf(psq - H, DT, s);

    float a0  = -r0 - TZO * v0 + u0;                // omega = 1
    float a1  = -r1 - TZO * v1 + u1;
    float vn0 = __builtin_fmaf(a0, DT, v0);
    float vn1 = __builtin_fmaf(a1, DT, v1);
    float rn0 = __builtin_fmaf(vn0, DT, r0);
    float rn1 = __builtin_fmaf(vn1, DT, r1);

    if (row < batch) {                              // post-WMMA predication OK
        float* o = out + (long)row * NX;
        __builtin_nontemporal_store(qn0, o + 0);
        __builtin_nontemporal_store(qn1, o + 1);
        __builtin_nontemporal_store(pn0, o + 2);
        __builtin_nontemporal_store(pn1, o + 3);
        __builtin_nontemporal_store(sn,  o + 4);
        __builtin_nontemporal_store(rn0, o + 5);
        __builtin_nontemporal_store(rn1, o + 6);
        __builtin_nontemporal_store(vn0, o + 7);
        __builtin_nontemporal_store(vn1, o + 8);
    }
}

extern "C" void kernel_launch(void* const* d_in, const int* in_sizes, int n_in,
                              void* d_out, int out_size, void* d_ws, size_t ws_size,
                              hipStream_t stream) {
    const float* x_int = (const float*)d_in[0];
    const float* u_t   = (const float*)d_in[1];
    const float* W1    = (const float*)d_in[2];
    const float* b1    = (const float*)d_in[3];
    const float* W2    = (const float*)d_in[4];
    const float* b2    = (const float*)d_in[5];
    float* out = (float*)d_out;

    const int batch = in_sizes[0] / NX;             // 4,194,304
    const int threads = 256;                        // 8 waves/block, 256 rows
    const int blocks  = (batch + threads - 1) / threads;
    resonant_step_kernel<<<blocks, threads, 0, stream>>>(
        x_int, u_t, W1, b1, W2, b2, out, batch);
}